// ODEVAE_34626026340478
// MI455X (gfx1250) — compile-verified
//
#include <hip/hip_runtime.h>
#include <hip/hip_bf16.h>
#include <math.h>

// ---------------------------------------------------------------------------
// ODE-VAE for MI455X (gfx1250): f32 WMMA 16x16x4 everywhere.
//   B=256, L=200, N=512, D=64, H=128
// ---------------------------------------------------------------------------

typedef __attribute__((ext_vector_type(2))) float v2f;
typedef __attribute__((ext_vector_type(8))) float v8f;

#define BB 256
#define LL 200
#define NN 512
#define DD 64
#define HH 128

// ---- WMMA f32 16x16x4 fragment helpers -------------------------------------
// A (MxK) row-major: lane<16 holds row M=lane, K = {k0,k0+1};
//                    lane>=16 holds row M=lane-16, K = {k0+2,k0+3}.
__device__ __forceinline__ v2f load_a_frag(const float* A, int lda, int m0, int k0, int lane) {
    int half = lane >> 4, r = lane & 15;
    const float* p = A + (long)(m0 + r) * lda + k0 + 2 * half;
    v2f a; a.x = p[0]; a.y = p[1];
    return a;
}
// B (KxN) row-major: lane<16 col N=lane, K = {k0,k0+1}; lane>=16: K = {k0+2,k0+3}.
__device__ __forceinline__ v2f load_b_frag(const float* Bm, int ldb, int k0, int n0, int lane) {
    int half = lane >> 4, r = lane & 15;
    const float* p = Bm + (long)(k0 + 2 * half) * ldb + n0 + r;
    v2f b; b.x = p[0]; b.y = p[ldb];
    return b;
}

__device__ __forceinline__ v8f wmma4(v2f a, v2f b, v8f c) {
    return __builtin_amdgcn_wmma_f32_16x16x4_f32(
        /*neg_a=*/false, a, /*neg_b=*/false, b,
        /*c_mod=*/(short)0, c, /*reuse_a=*/false, /*reuse_b=*/false);
}

// C/D layout: VGPR i -> row m0 + i + 8*(lane>=16), col n0 + (lane&15).
__device__ __forceinline__ void store_bias_relu(float* Dm, int ldd, int m0, int n0,
                                                int lane, v8f c, const float* bias) {
    int half = lane >> 4, r = lane & 15;
    float b = bias[n0 + r];
#pragma unroll
    for (int i = 0; i < 8; ++i) {
        float v = c[i] + b;
        Dm[(long)(m0 + i + 8 * half) * ldd + n0 + r] = v > 0.f ? v : 0.f;
    }
}
__device__ __forceinline__ void store_bias_silu(float* Dm, int ldd, int m0, int n0,
                                                int lane, v8f c, const float* bias) {
    int half = lane >> 4, r = lane & 15;
    float b = bias[n0 + r];
#pragma unroll
    for (int i = 0; i < 8; ++i) {
        float v = c[i] + b;
        Dm[(long)(m0 + i + 8 * half) * ldd + n0 + r] = v / (1.f + __expf(-v));
    }
}
__device__ __forceinline__ void store_bias(float* Dm, int ldd, int m0, int n0,
                                           int lane, v8f c, const float* bias) {
    int half = lane >> 4, r = lane & 15;
    float b = bias[n0 + r];
#pragma unroll
    for (int i = 0; i < 8; ++i)
        Dm[(long)(m0 + i + 8 * half) * ldd + n0 + r] = c[i] + b;
}

// Full K-loop 16x16 tile GEMM (A, B may live in LDS or global).
__device__ __forceinline__ v8f gemm_tile(const float* A, int lda, const float* Bw, int ldb,
                                         int m0, int n0, int K, int lane) {
    v8f acc = {};
    for (int k = 0; k < K; k += 4) {
        v2f a = load_a_frag(A, lda, m0, k, lane);
        v2f b = load_b_frag(Bw, ldb, k, n0, lane);
        acc = wmma4(a, b, acc);
    }
    return acc;
}

// ---------------------------------------------------------------------------
// Kernel 1: encoder.  16 workgroups x 256 threads; each owns 16 batch rows.
// ---------------------------------------------------------------------------
__global__ void __launch_bounds__(256)
encoder_kernel(const float* __restrict__ x_seq, const float* __restrict__ eps,
               const float* __restrict__ e_w1, const float* __restrict__ e_b1,
               const float* __restrict__ e_w2, const float* __restrict__ e_b2,
               const float* __restrict__ e_w3, const float* __restrict__ e_b3,
               const float* __restrict__ mu_w, const float* __restrict__ mu_b,
               const float* __restrict__ lv_w, const float* __restrict__ lv_b,
               float* __restrict__ mu_out, float* __restrict__ lv_out,
               float* __restrict__ ztraj) {
    __shared__ float xs[16 * 512];
    __shared__ float h1[16 * 512];
    __shared__ float h2[16 * 256];
    __shared__ float h3[16 * 128];
    __shared__ float mus[16 * 64];
    __shared__ float lvs[16 * 64];

    int tid = threadIdx.x, w = tid >> 5, lane = tid & 31;
    int m0g = blockIdx.x * 16;

    for (int i = tid; i < 16 * 512; i += 256) {
        int r = i >> 9, c = i & 511;
        xs[i] = x_seq[(long)(m0g + r) * (LL * NN) + c];   // x_seq[b, 0, :]
    }
    __syncthreads();

    for (int ct = w; ct < 32; ct += 8) {                  // 512 -> 512
        v8f a = gemm_tile(xs, 512, e_w1, 512, 0, ct * 16, 512, lane);
        store_bias_relu(h1, 512, 0, ct * 16, lane, a, e_b1);
    }
    __syncthreads();
    for (int ct = w; ct < 16; ct += 8) {                  // 512 -> 256
        v8f a = gemm_tile(h1, 512, e_w2, 256, 0, ct * 16, 512, lane);
        store_bias_relu(h2, 256, 0, ct * 16, lane, a, e_b2);
    }
    __syncthreads();
    {                                                     // 256 -> 128 (8 tiles, 1/wave)
        int ct = w;
        v8f a = gemm_tile(h2, 256, e_w3, 128, 0, ct * 16, 256, lane);
        store_bias_relu(h3, 128, 0, ct * 16, lane, a, e_b3);
    }
    __syncthreads();
    if (w < 4) {                                          // 128 -> 64 (mu)
        int ct = w;
        v8f a = gemm_tile(h3, 128, mu_w, 64, 0, ct * 16, 128, lane);
        store_bias(mus, 64, 0, ct * 16, lane, a, mu_b);
    } else {                                              // 128 -> 64 (logvar)
        int ct = w - 4;
        v8f a = gemm_tile(h3, 128, lv_w, 64, 0, ct * 16, 128, lane);
        store_bias(lvs, 64, 0, ct * 16, lane, a, lv_b);
    }
    __syncthreads();

    for (int i = tid; i < 16 * 64; i += 256) {
        int r = i >> 6, c = i & 63;
        long gb = m0g + r;
        float m = mus[i], l = lvs[i];
        mu_out[gb * DD + c] = m;
        lv_out[gb * DD + c] = l;
        ztraj[gb * (long)(LL * DD) + c] = m + __expf(0.5f * l) * eps[gb * DD + c];
    }
}

// ---------------------------------------------------------------------------
// Kernel 2: RK4 Neural-ODE scan.  Persistent: 16 WGs x 256 thr; weights in LDS.
// Thread tid owns z elements [tid*4 .. tid*4+3]  (row tid>>4, cols (tid&15)*4..)
// so layernorm reduces inside 16-lane shuffle groups and fuses with RK4.
// ---------------------------------------------------------------------------
__global__ void __launch_bounds__(256)
ode_kernel(const float* __restrict__ tvec,
           const float* __restrict__ f1_w1, const float* __restrict__ f1_b1,
           const float* __restrict__ f1_w2, const float* __restrict__ f1_b2,
           const float* __restrict__ f2_w, const float* __restrict__ f2_b,
           const float* __restrict__ ln_g, const float* __restrict__ ln_b,
           float* __restrict__ ztraj) {
    __shared__ float w2s[128 * 128];   // f1_w2 (64 KB)
    __shared__ float w3s[128 * 64];    // f2_w  (32 KB)
    __shared__ float b1s[128], b2s[128], b3s[64], lgs[64], lbs[64];
    __shared__ float dts[LL];          // per-step dt, preloaded
    __shared__ float ze[16 * 64];      // z evaluation point (A-matrix for layer 1)
    __shared__ float h1s[16 * 128];
    __shared__ float h2s[16 * 128];
    __shared__ float dzs[16 * 64];

    int tid = threadIdx.x, w = tid >> 5, lane = tid & 31;
    int m0g = blockIdx.x * 16;

    for (int i = tid; i < 128 * 128; i += 256) w2s[i] = f1_w2[i];
    for (int i = tid; i < 128 * 64; i += 256)  w3s[i] = f2_w[i];
    if (tid < 128) { b1s[tid] = f1_b1[tid]; b2s[tid] = f1_b2[tid]; }
    if (tid < 64)  { b3s[tid] = f2_b[tid]; lgs[tid] = ln_g[tid]; lbs[tid] = ln_b[tid]; }
    if (tid < LL - 1) dts[tid] = tvec[tid + 1] - tvec[tid];

    const int zr = tid >> 4;             // batch row within tile (0..15)
    const int zc = (tid & 15) * 4;       // first of 4 owned columns
    float zreg[4];                       // z_base lives in registers
#pragma unroll
    for (int j = 0; j < 4; ++j) {
        float v = ztraj[(long)(m0g + zr) * (LL * DD) + zc + j];   // z0
        zreg[j] = v;
        ze[zr * 64 + zc + j] = v;
    }
    __syncthreads();

    float ka[4];  // RK4 accumulator (registers)

    for (int s = 0; s < LL - 1; ++s) {
        float dt = dts[s];
#pragma unroll 1
        for (int stage = 0; stage < 4; ++stage) {
            {   // h1 = silu(ze @ f1_w1 + b1): 8 col-tiles, one per wave
                v8f a = gemm_tile(ze, 64, f1_w1, 128, 0, w * 16, 64, lane);
                store_bias_silu(h1s, 128, 0, w * 16, lane, a, b1s);
            }
            __syncthreads();
            {   // h2 = silu(h1 @ f1_w2 + b2)
                v8f a = gemm_tile(h1s, 128, w2s, 128, 0, w * 16, 128, lane);
                store_bias_silu(h2s, 128, 0, w * 16, lane, a, b2s);
            }
            __syncthreads();
            if (w < 4) {  // dz_raw = h2 @ f2_w + b3 : 4 col-tiles
                v8f a = gemm_tile(h2s, 128, w3s, 64, 0, w * 16, 128, lane);
                store_bias(dzs, 64, 0, w * 16, lane, a, b3s);
            }
            __syncthreads();
            // ---- fused layernorm (16-lane shuffle reduction) + RK4 combine ----
            float d0 = dzs[zr * 64 + zc + 0];
            float d1 = dzs[zr * 64 + zc + 1];
            float d2 = dzs[zr * 64 + zc + 2];
            float d3 = dzs[zr * 64 + zc + 3];
            float ssum = d0 + d1 + d2 + d3;
            ssum += __shfl_xor(ssum, 1);
            ssum += __shfl_xor(ssum, 2);
            ssum += __shfl_xor(ssum, 4);
            ssum += __shfl_xor(ssum, 8);
            float mean = ssum * (1.f / 64.f);
            float t0 = d0 - mean, t1 = d1 - mean, t2 = d2 - mean, t3 = d3 - mean;
            float q = t0 * t0 + t1 * t1 + t2 * t2 + t3 * t3;
            q += __shfl_xor(q, 1);
            q += __shfl_xor(q, 2);
            q += __shfl_xor(q, 4);
            q += __shfl_xor(q, 8);
            float inv = rsqrtf(q * (1.f / 64.f) + 1e-5f);
            float dn[4];
            dn[0] = t0 * inv * lgs[zc + 0] + lbs[zc + 0];
            dn[1] = t1 * inv * lgs[zc + 1] + lbs[zc + 1];
            dn[2] = t2 * inv * lgs[zc + 2] + lbs[zc + 2];
            dn[3] = t3 * inv * lgs[zc + 3] + lbs[zc + 3];
#pragma unroll
            for (int j = 0; j < 4; ++j) {
                int i = zr * 64 + zc + j;
                if (stage == 0)      { ka[j] = dn[j];        ze[i] = zreg[j] + 0.5f * dt * dn[j]; }
                else if (stage == 1) { ka[j] += 2.f * dn[j]; ze[i] = zreg[j] + 0.5f * dt * dn[j]; }
                else if (stage == 2) { ka[j] += 2.f * dn[j]; ze[i] = zreg[j] + dt * dn[j]; }
                else {
                    ka[j] += dn[j];
                    float zn = zreg[j] + dt * (1.f / 6.f) * ka[j];
                    zreg[j] = zn; ze[i] = zn;
                    ztraj[(long)(m0g + zr) * (LL * DD) + (long)(s + 1) * DD + zc + j] = zn;
                }
            }
            __syncthreads();
        }
    }
}

// ---------------------------------------------------------------------------
// Kernel 3: zdiff = (z[:,1:]-z[:,:-1]) / dt
// ---------------------------------------------------------------------------
__global__ void __launch_bounds__(256)
zdiff_kernel(const float* __restrict__ ztraj, const float* __restrict__ tvec,
             float* __restrict__ zdiff) {
    long idx = (long)blockIdx.x * 256 + threadIdx.x;
    if (idx >= (long)BB * (LL - 1) * DD) return;
    int c = (int)(idx & 63);
    long rem = idx >> 6;
    int t = (int)(rem % (LL - 1));
    long b = rem / (LL - 1);
    float dt = tvec[t + 1] - tvec[t];
    float a = ztraj[b * (LL * DD) + (long)t * DD + c];
    float e = ztraj[b * (LL * DD) + (long)(t + 1) * DD + c];
    zdiff[idx] = (e - a) / dt;
}

// ---------------------------------------------------------------------------
// Kernel 4: decoder. 800 WGs x 256 thr; 64-row tile (272 KB LDS of 320 KB/WGP).
// Each B-fragment is reused across 4 row sub-tiles -> 0.5 vmem-instr per WMMA,
// 2x less L2 weight traffic than 32-row tiles. Weight stream prefetched.
// ---------------------------------------------------------------------------
__global__ void __launch_bounds__(256)
decoder_kernel(const float* __restrict__ ztraj,
               const float* __restrict__ d_w1, const float* __restrict__ d_b1,
               const float* __restrict__ d_w2, const float* __restrict__ d_b2,
               const float* __restrict__ d_w3, const float* __restrict__ d_b3,
               float* __restrict__ xhat) {
    __shared__ float zt[64 * 64];     //  16 KB
    __shared__ float h1[64 * 512];    // 128 KB
    __shared__ float h2[64 * 512];    // 128 KB

    int tid = threadIdx.x, w = tid >> 5, lane = tid & 31;
    long r0 = (long)blockIdx.x * 64;  // rows of flattened (B*L, 64)

    for (int i = tid; i < 64 * 64; i += 256) zt[i] = ztraj[r0 * DD + i];
    __syncthreads();

    // layer 1: 64 -> 512, relu
    for (int ct = w; ct < 32; ct += 8) {
        int n0 = ct * 16;
        v8f acc[4] = {{}, {}, {}, {}};
        for (int k = 0; k < 64; k += 4) {
            v2f b = load_b_frag(d_w1, 512, k, n0, lane);
#pragma unroll
            for (int rt = 0; rt < 4; ++rt) {
                v2f a = load_a_frag(zt, 64, rt * 16, k, lane);
                acc[rt] = wmma4(a, b, acc[rt]);
            }
        }
#pragma unroll
        for (int rt = 0; rt < 4; ++rt)
            store_bias_relu(h1, 512, rt * 16, n0, lane, acc[rt], d_b1);
    }
    __syncthreads();

    // layer 2: 512 -> 512, relu
    for (int ct = w; ct < 32; ct += 8) {
        int n0 = ct * 16;
        v8f acc[4] = {{}, {}, {}, {}};
        for (int k = 0; k < 512; k += 4) {
            if ((k & 63) == 0 && k + 128 < 512)
                __builtin_prefetch(&d_w2[(long)(k + 128) * 512 + n0 + lane], 0, 0);
            v2f b = load_b_frag(d_w2, 512, k, n0, lane);
#pragma unroll
            for (int rt = 0; rt < 4; ++rt) {
                v2f a = load_a_frag(h1, 512, rt * 16, k, lane);
                acc[rt] = wmma4(a, b, acc[rt]);
            }
        }
#pragma unroll
        for (int rt = 0; rt < 4; ++rt)
            store_bias_relu(h2, 512, rt * 16, n0, lane, acc[rt], d_b2);
    }
    __syncthreads();

    // layer 3: 512 -> 512, linear, store to global xhat
    float* xo = xhat + r0 * NN;
    for (int ct = w; ct < 32; ct += 8) {
        int n0 = ct * 16;
        v8f acc[4] = {{}, {}, {}, {}};
        for (int k = 0; k < 512; k += 4) {
            if ((k & 63) == 0 && k + 128 < 512)
                __builtin_prefetch(&d_w3[(long)(k + 128) * 512 + n0 + lane], 0, 0);
            v2f b = load_b_frag(d_w3, 512, k, n0, lane);
#pragma unroll
            for (int rt = 0; rt < 4; ++rt) {
                v2f a = load_a_frag(h2, 512, rt * 16, k, lane);
                acc[rt] = wmma4(a, b, acc[rt]);
            }
        }
#pragma unroll
        for (int rt = 0; rt < 4; ++rt)
            store_bias(xo, 512, rt * 16, n0, lane, acc[rt], d_b3);
    }
}

// ---------------------------------------------------------------------------
extern "C" void kernel_launch(void* const* d_in, const int* in_sizes, int n_in,
                              void* d_out, int out_size, void* d_ws, size_t ws_size,
                              hipStream_t stream) {
    const float* x_seq = (const float*)d_in[0];
    const float* tvec  = (const float*)d_in[1];
    const float* eps   = (const float*)d_in[2];
    const float* e_w1  = (const float*)d_in[3];
    const float* e_b1  = (const float*)d_in[4];
    const float* e_w2  = (const float*)d_in[5];
    const float* e_b2  = (const float*)d_in[6];
    const float* e_w3  = (const float*)d_in[7];
    const float* e_b3  = (const float*)d_in[8];
    const float* mu_w  = (const float*)d_in[9];
    const float* mu_b  = (const float*)d_in[10];
    const float* lv_w  = (const float*)d_in[11];
    const float* lv_b  = (const float*)d_in[12];
    const float* f1_w1 = (const float*)d_in[13];
    const float* f1_b1 = (const float*)d_in[14];
    const float* f1_w2 = (const float*)d_in[15];
    const float* f1_b2 = (const float*)d_in[16];
    const float* f2_w  = (const float*)d_in[17];
    const float* f2_b  = (const float*)d_in[18];
    const float* ln_g  = (const float*)d_in[19];
    const float* ln_b  = (const float*)d_in[20];
    const float* d_w1  = (const float*)d_in[21];
    const float* d_b1  = (const float*)d_in[22];
    const float* d_w2  = (const float*)d_in[23];
    const float* d_b2  = (const float*)d_in[24];
    const float* d_w3  = (const float*)d_in[25];
    const float* d_b3  = (const float*)d_in[26];

    float* out = (float*)d_out;
    const long MU   = (long)BB * LL * NN;           // 26,214,400
    const long LV   = MU + (long)BB * DD;
    const long ZT   = LV + (long)BB * DD;
    const long ZD   = ZT + (long)BB * LL * DD;
    float* xhat = out;
    float* mu_o = out + MU;
    float* lv_o = out + LV;
    float* zt   = out + ZT;
    float* zd   = out + ZD;

    encoder_kernel<<<16, 256, 0, stream>>>(x_seq, eps, e_w1, e_b1, e_w2, e_b2,
                                           e_w3, e_b3, mu_w, mu_b, lv_w, lv_b,
                                           mu_o, lv_o, zt);
    ode_kernel<<<16, 256, 0, stream>>>(tvec, f1_w1, f1_b1, f1_w2, f1_b2,
                                       f2_w, f2_b, ln_g, ln_b, zt);
    zdiff_kernel<<<12736, 256, 0, stream>>>(zt, tvec, zd);
    decoder_kernel<<<800, 256, 0, stream>>>(zt, d_w1, d_b1, d_w2, d_b2,
                                            d_w3, d_b3, xhat);
}